// att_conv_LN_normalize_12489764897310
// MI455X (gfx1250) — compile-verified
//
#include <hip/hip_runtime.h>
#include <hip/hip_bf16.h>

// ---------------------------------------------------------------------------
// Fused LN -> {k,v,q} 1x1 convs (64->32, fp32 WMMA 16x16x4) -> pair-softmax
// attention gate -> conv32->16 +LN+relu -> conv16->8 +LN+relu -> conv8->16.
// MI455X / gfx1250, wave32. Memory-bound target: ~300MB @ 23.3TB/s ~= 13us.
//
// Round-3: cap register usage to <=256 VGPRs (amdgpu_waves_per_eu(4)) to kill
// s_set_vgpr_msb traffic and run 4 waves/SIMD; main-loop unroll reduced to 2
// to shrink the scheduler's in-flight operand window; prefetch block trimmed.
// ---------------------------------------------------------------------------

typedef __attribute__((ext_vector_type(2))) float v2f;
typedef __attribute__((ext_vector_type(8))) float v8f;

#define HW    65536      // 256*256 pixels per image plane
#define NPB   4096       // 16-pixel groups per batch image

__device__ __forceinline__ v8f wmma4(v2f a, v2f b, v8f c) {
  // D(16x16,f32) = A(16x4,f32) x B(4x16,f32) + C
  return __builtin_amdgcn_wmma_f32_16x16x4_f32(
      /*neg_a=*/false, a, /*neg_b=*/false, b,
      /*c_mod=*/(short)0, c, /*reuse_a=*/false, /*reuse_b=*/false);
}

__global__ __launch_bounds__(256) __attribute__((amdgpu_waves_per_eu(4)))
void fused_attconv_ln_kernel(
    const float* __restrict__ kv,   const float* __restrict__ qin,
    const float* __restrict__ ln_g, const float* __restrict__ ln_b,
    const float* __restrict__ Wk,   const float* __restrict__ bk,
    const float* __restrict__ Wv,   const float* __restrict__ bv,
    const float* __restrict__ Wq,   const float* __restrict__ bq,
    const float* __restrict__ scale,
    const float* __restrict__ W1,   const float* __restrict__ g1, const float* __restrict__ b1,
    const float* __restrict__ W2,   const float* __restrict__ g2, const float* __restrict__ b2,
    const float* __restrict__ W3,   const float* __restrict__ b3,
    float* __restrict__ out)
{
  // Weights pre-swizzled to WMMA A-operand layout: idx = t*1024 + kb*64 + lane*2 + j,
  // holding W[m][c] with m = t*16 + (lane&15), c = 4*kb + 2*(lane>>4) + j.
  __shared__ __attribute__((aligned(16))) float wAk[2048];
  __shared__ __attribute__((aligned(16))) float wAv[2048];
  __shared__ __attribute__((aligned(16))) float wAq[2048];
  __shared__ __attribute__((aligned(16))) float wA1[512];
  __shared__ __attribute__((aligned(16))) float wA2[256];
  __shared__ __attribute__((aligned(16))) float wA3[128];
  __shared__ float bkp[32], bvp[32], bqp[32];
  __shared__ float rkp[32], rvp[32];               // LN-folded weight rowsums
  __shared__ float sscale[16], sg1[16], sb1[16], sg2[16], sb2[16], sb3[16];
  __shared__ __attribute__((aligned(16))) float scr[8][512];  // per-wave pivot scratch

  const int tid = threadIdx.x;

  // ---- cooperative weight prep (LN affine folded into Wk/Wv and biases) ----
  for (int idx = tid; idx < 2048; idx += 256) {
    int j = idx & 1, lane = (idx >> 1) & 31, kb = (idx >> 6) & 15, t = idx >> 10;
    int m = t * 16 + (lane & 15);
    int c = 4 * kb + 2 * (lane >> 4) + j;
    float gc = ln_g[c];
    wAk[idx] = Wk[m * 64 + c] * gc;
    wAv[idx] = Wv[m * 64 + c] * gc;
    wAq[idx] = Wq[m * 64 + c];
  }
  for (int idx = tid; idx < 512; idx += 256) {
    int j = idx & 1, lane = (idx >> 1) & 31, kb = (idx >> 6) & 7;
    int m = lane & 15;
    int c = 4 * kb + 2 * (lane >> 4) + j;
    wA1[idx] = W1[m * 32 + c];
  }
  if (tid < 256) {
    int idx = tid;
    int j = idx & 1, lane = (idx >> 1) & 31, kb = (idx >> 6) & 3;
    int m = lane & 15;
    int c = 4 * kb + 2 * (lane >> 4) + j;
    wA2[idx] = (m < 8) ? W2[m * 16 + c] : 0.0f;   // pad M 8->16 with zero rows
  }
  if (tid < 128) {
    int idx = tid;
    int j = idx & 1, lane = (idx >> 1) & 31, kb = (idx >> 6) & 1;
    int m = lane & 15;
    int c = 4 * kb + 2 * (lane >> 4) + j;
    wA3[idx] = W3[m * 8 + c];
  }
  if (tid < 32) {
    float sk = bk[tid], sv = bv[tid], rk = 0.f, rv = 0.f;
    for (int c = 0; c < 64; ++c) {
      float wkc = Wk[tid * 64 + c], wvc = Wv[tid * 64 + c], gc = ln_g[c];
      sk += wkc * ln_b[c];
      sv += wvc * ln_b[c];
      rk += wkc * gc;                 // rowsum of LN-folded weights
      rv += wvc * gc;
    }
    bkp[tid] = sk; bvp[tid] = sv; rkp[tid] = rk; rvp[tid] = rv; bqp[tid] = bq[tid];
  }
  if (tid < 16) {
    sscale[tid] = scale[tid];
    sg1[tid] = g1[tid]; sb1[tid] = b1[tid];
    sg2[tid] = (tid < 8) ? g2[tid] : 0.0f;
    sb2[tid] = (tid < 8) ? b2[tid] : 0.0f;
    sb3[tid] = b3[tid];
  }
  __syncthreads();

  const int lane = tid & 31;
  const int wv   = tid >> 5;
  const int h    = lane >> 4;   // lane half: selects K/M sub-block in WMMA layouts
  const int pl   = lane & 15;   // pixel within 16-pixel group (WMMA N index)
  float* myscr = &scr[wv][0];
  const int wgid = blockIdx.x * 8 + wv;   // 4096 waves, 8 groups each = 32768 groups

  #pragma unroll 1
  for (int g = 0; g < 8; ++g) {
    const int  G   = wgid * 8 + g;
    const int  b   = G >> 12;                       // 4096 groups per image
    const long pin = (long)(G & (NPB - 1)) * 16;
    const float* xb = kv  + (long)b * 64 * HW + pin + (long)h * 2 * HW + pl;
    const float* qb = qin + (long)b * 64 * HW + pin + (long)h * 2 * HW + pl;
    float*       ob = out + (long)b * 16 * HW + pin + (long)h * 8 * HW + pl;

    // light prefetch of the next group (global_prefetch_b8)
    if (g + 1 < 8) {
      __builtin_prefetch(xb + 16, 0, 0);
      __builtin_prefetch(qb + 16, 0, 0);
    }

    // ------- fused streaming GEMMs: qr = Wq q + bq, kraw = Wk' x, vraw = Wv' x
    // (LN applied post-hoc: Wk'.(x-mu)rs = rs.(kraw - mu.Rk)); x/q never resident
    v8f qr0, qr1, k0, k1, u0, u1;
    #pragma unroll
    for (int i = 0; i < 8; ++i) {
      qr0[i] = bqp[i + 8 * h]; qr1[i] = bqp[16 + i + 8 * h];
      k0[i] = 0.f; k1[i] = 0.f; u0[i] = 0.f; u1[i] = 0.f;
    }
    float sm = 0.f, sq = 0.f;
    #pragma unroll 2
    for (int kb = 0; kb < 16; ++kb) {
      v2f bqv, bxv;
      bqv.x = qb[(4 * kb + 0) * HW];
      bqv.y = qb[(4 * kb + 1) * HW];
      bxv.x = xb[(4 * kb + 0) * HW];
      bxv.y = xb[(4 * kb + 1) * HW];
      sm += bxv.x + bxv.y;
      sq += bxv.x * bxv.x + bxv.y * bxv.y;
      v2f aq0 = *(const v2f*)&wAq[kb * 64 + lane * 2];
      v2f aq1 = *(const v2f*)&wAq[1024 + kb * 64 + lane * 2];
      v2f ak0 = *(const v2f*)&wAk[kb * 64 + lane * 2];
      v2f ak1 = *(const v2f*)&wAk[1024 + kb * 64 + lane * 2];
      v2f av0 = *(const v2f*)&wAv[kb * 64 + lane * 2];
      v2f av1 = *(const v2f*)&wAv[1024 + kb * 64 + lane * 2];
      qr0 = wmma4(aq0, bqv, qr0);     // 6 independent accumulation chains
      qr1 = wmma4(aq1, bqv, qr1);
      k0  = wmma4(ak0, bxv, k0);
      k1  = wmma4(ak1, bxv, k1);
      u0  = wmma4(av0, bxv, u0);
      u1  = wmma4(av1, bxv, u1);
    }
    // per-pixel LN stats (other 32 channels live in partner lane half)
    sm += __shfl_xor(sm, 16, 32);
    sq += __shfl_xor(sq, 16, 32);
    float mu  = sm * (1.f / 64.f);
    float var = sq * (1.f / 64.f) - mu * mu;
    float rs  = rsqrtf(var + 1e-5f);

    // ------- LN fixup + paired softmax gate + s = v*att -> scratch --------
    // pair (c, c+16) sits at the same (reg,lane) slot of tile0/tile1
    #pragma unroll
    for (int i = 0; i < 8; ++i) {
      int m0 = i + 8 * h, m1 = 16 + m0;
      float kk0 = rs * (k0[i] - mu * rkp[m0]) + bkp[m0];
      float kk1 = rs * (k1[i] - mu * rkp[m1]) + bkp[m1];
      float vv0 = rs * (u0[i] - mu * rvp[m0]) + bvp[m0];
      float vv1 = rs * (u1[i] - mu * rvp[m1]) + bvp[m1];
      float sc  = sscale[m0];                        // scale index = oc % 16
      float d   = (kk1 * qr1[i] - kk0 * qr0[i]) * sc;
      float p0  = 1.f / (1.f + __expf(d));           // stable 2-way softmax
      float p1  = 1.f / (1.f + __expf(-d));
      myscr[m0 * 16 + pl] = vv0 * p0;                // D layout -> [c][p] in LDS
      myscr[m1 * 16 + pl] = vv1 * p1;
    }

    // ---------------- conv 32->16, LN16, relu -----------------------------
    v8f c1 = {0.f, 0.f, 0.f, 0.f, 0.f, 0.f, 0.f, 0.f};
    #pragma unroll
    for (int kb = 0; kb < 8; ++kb) {
      v2f bs;
      bs.x = myscr[(4 * kb + 2 * h) * 16 + pl];
      bs.y = myscr[(4 * kb + 2 * h + 1) * 16 + pl];
      v2f a = *(const v2f*)&wA1[kb * 64 + lane * 2];
      c1 = wmma4(a, bs, c1);
    }
    float lm = 0.f, lq = 0.f;
    #pragma unroll
    for (int i = 0; i < 8; ++i) { lm += c1[i]; lq += c1[i] * c1[i]; }
    lm += __shfl_xor(lm, 16, 32);
    lq += __shfl_xor(lq, 16, 32);
    mu  = lm * (1.f / 16.f);
    var = lq * (1.f / 16.f) - mu * mu;
    rs  = rsqrtf(var + 1e-5f);
    #pragma unroll
    for (int i = 0; i < 8; ++i) {
      int m = i + 8 * h;
      float hv = fmaxf((c1[i] - mu) * rs * sg1[m] + sb1[m], 0.f);
      myscr[m * 16 + pl] = hv;              // 16x16 tile back to [c][p]
    }

    // ---------------- conv 16->8 (M padded), LN8, relu --------------------
    v8f c2 = {0.f, 0.f, 0.f, 0.f, 0.f, 0.f, 0.f, 0.f};
    #pragma unroll
    for (int kb = 0; kb < 4; ++kb) {
      v2f bs;
      bs.x = myscr[(4 * kb + 2 * h) * 16 + pl];
      bs.y = myscr[(4 * kb + 2 * h + 1) * 16 + pl];
      v2f a = *(const v2f*)&wA2[kb * 64 + lane * 2];
      c2 = wmma4(a, bs, c2);
    }
    // channels 0..7 live entirely in h==0 lanes -> lane-local LN; h==1 lanes
    // hold exact zeros (zero-padded weight rows) and padded g2/b2=0 -> emit 0.
    lm = 0.f; lq = 0.f;
    #pragma unroll
    for (int i = 0; i < 8; ++i) { lm += c2[i]; lq += c2[i] * c2[i]; }
    mu  = lm * (1.f / 8.f);
    var = lq * (1.f / 8.f) - mu * mu;
    rs  = rsqrtf(var + 1e-5f);
    #pragma unroll
    for (int i = 0; i < 8; ++i) {
      int m = i + 8 * h;
      float hv = fmaxf((c2[i] - mu) * rs * sg2[m] + sb2[m], 0.f);
      myscr[m * 16 + pl] = hv;              // only c<8 is read below
    }

    // ---------------- conv 8->16 + b3, store output -----------------------
    v8f c3;
    #pragma unroll
    for (int i = 0; i < 8; ++i) c3[i] = sb3[i + 8 * h];
    #pragma unroll
    for (int kb = 0; kb < 2; ++kb) {
      v2f bs;
      bs.x = myscr[(4 * kb + 2 * h) * 16 + pl];
      bs.y = myscr[(4 * kb + 2 * h + 1) * 16 + pl];
      v2f a = *(const v2f*)&wA3[kb * 64 + lane * 2];
      c3 = wmma4(a, bs, c3);
    }
    #pragma unroll
    for (int i = 0; i < 8; ++i) ob[i * HW] = c3[i];
  }
}

extern "C" void kernel_launch(void* const* d_in, const int* in_sizes, int n_in,
                              void* d_out, int out_size, void* d_ws, size_t ws_size,
                              hipStream_t stream) {
  (void)in_sizes; (void)n_in; (void)out_size; (void)d_ws; (void)ws_size;
  const float* kv    = (const float*)d_in[0];
  const float* q     = (const float*)d_in[1];
  const float* ln_g  = (const float*)d_in[2];
  const float* ln_b  = (const float*)d_in[3];
  const float* Wk    = (const float*)d_in[4];
  const float* bk    = (const float*)d_in[5];
  const float* Wv    = (const float*)d_in[6];
  const float* bv    = (const float*)d_in[7];
  const float* Wq    = (const float*)d_in[8];
  const float* bq    = (const float*)d_in[9];
  const float* scale = (const float*)d_in[10];
  const float* W1    = (const float*)d_in[11];
  const float* g1    = (const float*)d_in[12];
  const float* b1    = (const float*)d_in[13];
  const float* W2    = (const float*)d_in[14];
  const float* g2    = (const float*)d_in[15];
  const float* b2    = (const float*)d_in[16];
  const float* W3    = (const float*)d_in[17];
  const float* b3    = (const float*)d_in[18];
  float* out = (float*)d_out;

  // 512 blocks * 8 waves * 8 groups * 16 pixels = 524288 pixels (exact cover)
  fused_attconv_ln_kernel<<<dim3(512), dim3(256), 0, stream>>>(
      kv, q, ln_g, ln_b, Wk, bk, Wv, bv, Wq, bq, scale,
      W1, g1, b1, W2, g2, b2, W3, b3, out);
}